// GATPerformancePrediction_42245298324234
// MI455X (gfx1250) — compile-verified
//
#include <hip/hip_runtime.h>
#include <math.h>

typedef __attribute__((ext_vector_type(2))) float v2f;
typedef __attribute__((ext_vector_type(8))) float v8f;

#define LRELU_SLOPE 0.2f

__device__ __forceinline__ float lrelu(float x){ return x > 0.f ? x : LRELU_SLOPE * x; }
__device__ __forceinline__ float elu1(float x){ return x > 0.f ? x : (expf(x) - 1.f); }

// order-preserving float<->uint encoding for unsigned atomicMax
__device__ __forceinline__ unsigned fenc(float f){
  unsigned u = __float_as_uint(f);
  return (u & 0x80000000u) ? ~u : (u | 0x80000000u);
}
__device__ __forceinline__ float fdec(unsigned e){
  return __uint_as_float((e & 0x80000000u) ? (e & 0x7fffffffu) : ~e);
}

__global__ void k_fill(float* p, float v, int n){
  int i = blockIdx.x * blockDim.x + threadIdx.x;
  if (i < n) p[i] = v;
}
__global__ void k_fillu(unsigned* p, unsigned v, int n){
  int i = blockIdx.x * blockDim.x + threadIdx.x;
  if (i < n) p[i] = v;
}

// h[n, 0:128] = emb[x[n,0], 0:128]
__global__ void k_embed(const int* __restrict__ x, const float* __restrict__ emb,
                        float* __restrict__ h, int N){
  int idx = blockIdx.x * blockDim.x + threadIdx.x;
  if (idx >= N * 128) return;
  int n = idx >> 7, k = idx & 127;
  h[idx] = emb[(size_t)x[n * 2] * 128 + k];
}

// out[N x 256] = A[N x D] * W[D x 256], f32 WMMA 16x16x4, one wave per 16x16 tile
__global__ void k_gemm_wmma(const float* __restrict__ A, const float* __restrict__ W,
                            float* __restrict__ out, int N, int D){
  int wave = (blockIdx.x * blockDim.x + threadIdx.x) >> 5;
  int rowTiles = (N + 15) >> 4;
  if (wave >= rowTiles * 16) return;           // uniform per wave
  int rt = wave >> 4;                           // 16 col tiles (256 cols)
  int ct = wave & 15;
  int row0 = rt << 4, col0 = ct << 4;
  int lane = threadIdx.x & 31;
  int mr   = lane & 15;                         // A: M index / B: N index
  int koff = (lane >> 4) << 1;                  // lanes 16-31 hold K=2,3
  int arow = row0 + mr; if (arow >= N) arow = N - 1;   // clamp (no divergence)
  const float* Arow = A + (size_t)arow * D;
  v8f c = {0.f,0.f,0.f,0.f,0.f,0.f,0.f,0.f};
  for (int k = 0; k < D; k += 4){
    v2f a, b;
    a.x = Arow[k + koff];
    a.y = Arow[k + koff + 1];
    b.x = W[(size_t)(k + koff)     * 256 + col0 + mr];
    b.y = W[(size_t)(k + koff + 1) * 256 + col0 + mr];
    c = __builtin_amdgcn_wmma_f32_16x16x4_f32(false, a, false, b, (short)0, c, false, false);
  }
  // C/D layout: VGPR r holds M = r (lanes 0-15) / M = 8+r (lanes 16-31), N = lane&15
  int mb = (lane >> 4) << 3;
  for (int r = 0; r < 8; ++r){
    int m = row0 + mb + r;
    if (m < N) out[(size_t)m * 256 + col0 + mr] = c[r];
  }
}

// per-(node, head) attention coefficients
__global__ void k_att(const float* __restrict__ hp, const float* __restrict__ as_,
                      const float* __restrict__ ad_, float* __restrict__ aS,
                      float* __restrict__ aD, int N){
  int idx = blockIdx.x * blockDim.x + threadIdx.x;
  if (idx >= N * 4) return;
  int n = idx >> 2, h = idx & 3;
  const float* row = hp + (size_t)n * 256 + h * 64;
  const float* s = as_ + h * 64;
  const float* d = ad_ + h * 64;
  float accS = 0.f, accD = 0.f;
  for (int ch = 0; ch < 64; ++ch){ float v = row[ch]; accS += v * s[ch]; accD += v * d[ch]; }
  aS[idx] = accS; aD[idx] = accD;
}

__device__ __forceinline__ void edge_nodes(const int* ei, int E, int i, int& s, int& d){
  if (i < E){ s = ei[i]; d = ei[E + i]; } else { s = i - E; d = i - E; }
}

__global__ void k_edge_max(const int* __restrict__ ei, int E, int Etot,
                           const float* __restrict__ aS, const float* __restrict__ aD,
                           unsigned* __restrict__ m, float* __restrict__ eV){
  int idx = blockIdx.x * blockDim.x + threadIdx.x;
  if (idx >= Etot * 4) return;
  int i = idx >> 2, h = idx & 3;
  int s, d; edge_nodes(ei, E, i, s, d);
  float e = lrelu(aS[s * 4 + h] + aD[d * 4 + h]);
  eV[idx] = e;
  atomicMax(&m[d * 4 + h], fenc(e));
}

__global__ void k_edge_exp(const int* __restrict__ ei, int E, int Etot,
                           const unsigned* __restrict__ m, float* __restrict__ eV,
                           float* __restrict__ ssum){
  int idx = blockIdx.x * blockDim.x + threadIdx.x;
  if (idx >= Etot * 4) return;
  int i = idx >> 2, h = idx & 3;
  int s, d; edge_nodes(ei, E, i, s, d);
  float p = expf(eV[idx] - fdec(m[d * 4 + h]));
  eV[idx] = p;
  atomicAdd(&ssum[d * 4 + h], p);
}

__global__ void k_edge_alpha(const int* __restrict__ ei, int E, int Etot,
                             const float* __restrict__ ssum, float* __restrict__ eV){
  int idx = blockIdx.x * blockDim.x + threadIdx.x;
  if (idx >= Etot * 4) return;
  int i = idx >> 2, h = idx & 3;
  int s, d; edge_nodes(ei, E, i, s, d);
  eV[idx] = eV[idx] / (ssum[d * 4 + h] + 1e-16f);
}

// acc[dst, h, c] += alpha[edge,h] * hp[src, h, c]   (256 lanes per edge)
__global__ void k_edge_msg(const int* __restrict__ ei, int E, long long total,
                           const float* __restrict__ eV, const float* __restrict__ hp,
                           float* __restrict__ acc){
  long long idx = (long long)blockIdx.x * blockDim.x + threadIdx.x;
  if (idx >= total) return;
  int i = (int)(idx >> 8), t = (int)(idx & 255), h = t >> 6;
  int s, d; edge_nodes(ei, E, i, s, d);
  atomicAdd(&acc[(size_t)d * 256 + t], eV[i * 4 + h] * hp[(size_t)s * 256 + t]);
}

// mean over heads + bias, elu -> new node features (N x 64)
__global__ void k_finalize(const float* __restrict__ acc, const float* __restrict__ bias,
                           float* __restrict__ hOut, int N){
  int idx = blockIdx.x * blockDim.x + threadIdx.x;
  if (idx >= N * 64) return;
  int n = idx >> 6, c = idx & 63;
  const float* a = acc + (size_t)n * 256;
  float v = 0.25f * (a[c] + a[64 + c] + a[128 + c] + a[192 + c]) + bias[c];
  hOut[idx] = elu1(v);
}

__global__ void k_pool(const float* __restrict__ h, const int* __restrict__ batch,
                       float* __restrict__ gs, float* __restrict__ gc, int N){
  int idx = blockIdx.x * blockDim.x + threadIdx.x;
  if (idx >= N * 64) return;
  int n = idx >> 6, c = idx & 63;
  int g = batch[n];
  atomicAdd(&gs[g * 64 + c], h[idx]);
  if (c == 0) atomicAdd(&gc[g], 1.f);
}

__global__ void k_pool_div(float* gs, const float* gc){
  int idx = blockIdx.x * blockDim.x + threadIdx.x;
  if (idx >= 64 * 64) return;
  gs[idx] = gs[idx] / fmaxf(gc[idx >> 6], 1.f);
}

// pooled(64x64) -> elu(@mW1+mb1) -> @mW2+mb2 -> @rW+rb ; single 1024-thread block
__global__ void k_mlp(const float* __restrict__ pooled, const float* __restrict__ mW1,
                      const float* __restrict__ mb1, const float* __restrict__ mW2,
                      const float* __restrict__ mb2, const float* __restrict__ rW,
                      const float* __restrict__ rb, float* __restrict__ out){
  __shared__ float P[64 * 64];
  __shared__ float R1[64 * 64];
  __shared__ float R2[64 * 64];
  int t = threadIdx.x;                       // 1024 threads
  for (int i = t; i < 4096; i += 1024) P[i] = pooled[i];
  __syncthreads();
  int j = t & 63, gq = t >> 6;               // gq in [0,16)
  for (int rep = 0; rep < 4; ++rep){
    int g = gq + rep * 16;
    float a = mb1[j];
    for (int k = 0; k < 64; ++k) a += P[g * 64 + k] * mW1[k * 64 + j];
    R1[g * 64 + j] = elu1(a);
  }
  __syncthreads();
  for (int rep = 0; rep < 4; ++rep){
    int g = gq + rep * 16;
    float a = mb2[j];
    for (int k = 0; k < 64; ++k) a += R1[g * 64 + k] * mW2[k * 64 + j];
    R2[g * 64 + j] = a;
  }
  __syncthreads();
  if (t < 64){
    float a = rb[0];
    for (int k = 0; k < 64; ++k) a += R2[t * 64 + k] * rW[k];
    out[t] = a;
  }
}

static inline int cdivll(long long a, long long b){ return (int)((a + b - 1) / b); }

extern "C" void kernel_launch(void* const* d_in, const int* in_sizes, int n_in,
                              void* d_out, int out_size, void* d_ws, size_t ws_size,
                              hipStream_t stream){
  const int*   x     = (const int*)d_in[0];
  const int*   ei    = (const int*)d_in[1];   // [2, E] row-major
  const int*   batch = (const int*)d_in[2];
  const float* emb   = (const float*)d_in[3];
  const float* Wl[3]  = {(const float*)d_in[4],  (const float*)d_in[8],  (const float*)d_in[12]};
  const float* asl[3] = {(const float*)d_in[5],  (const float*)d_in[9],  (const float*)d_in[13]};
  const float* adl[3] = {(const float*)d_in[6],  (const float*)d_in[10], (const float*)d_in[14]};
  const float* bl[3]  = {(const float*)d_in[7],  (const float*)d_in[11], (const float*)d_in[15]};
  const float* mW1 = (const float*)d_in[16];
  const float* mb1 = (const float*)d_in[17];
  const float* mW2 = (const float*)d_in[18];
  const float* mb2 = (const float*)d_in[19];
  const float* rW  = (const float*)d_in[20];
  const float* rb  = (const float*)d_in[21];

  const int N    = in_sizes[0] / 2;
  const int E    = in_sizes[1] / 2;
  const int Etot = E + N;

  float* ws = (float*)d_ws;
  size_t off = 0;
  float* h    = ws + off; off += (size_t)N * 128;
  float* hp   = ws + off; off += (size_t)N * 256;
  float* acc  = ws + off; off += (size_t)N * 256;
  float* aS   = ws + off; off += (size_t)N * 4;
  float* aD   = ws + off; off += (size_t)N * 4;
  float* eV   = ws + off; off += (size_t)Etot * 4;
  unsigned* mMax = (unsigned*)(ws + off); off += (size_t)N * 4;
  float* sSum = ws + off; off += (size_t)N * 4;
  float* gs   = ws + off; off += 64 * 64;
  float* gc   = ws + off; off += 64;
  (void)ws_size; (void)n_in; (void)out_size;

  const unsigned ENC_NEG_INF = 0x007FFFFFu;   // fenc(-inf)

  k_embed<<<cdivll((long long)N * 128, 256), 256, 0, stream>>>(x, emb, h, N);

  for (int L = 0; L < 3; ++L){
    int D = (L == 0) ? 128 : 64;
    k_fill <<<cdivll((long long)N * 256, 256), 256, 0, stream>>>(acc, 0.f, N * 256);
    k_fillu<<<cdivll((long long)N * 4,   256), 256, 0, stream>>>(mMax, ENC_NEG_INF, N * 4);
    k_fill <<<cdivll((long long)N * 4,   256), 256, 0, stream>>>(sSum, 0.f, N * 4);

    long long waves = (long long)((N + 15) / 16) * 16;   // one wave per 16x16 tile
    k_gemm_wmma<<<cdivll(waves * 32, 256), 256, 0, stream>>>(h, Wl[L], hp, N, D);

    k_att<<<cdivll((long long)N * 4, 256), 256, 0, stream>>>(hp, asl[L], adl[L], aS, aD, N);
    k_edge_max  <<<cdivll((long long)Etot * 4, 256), 256, 0, stream>>>(ei, E, Etot, aS, aD, mMax, eV);
    k_edge_exp  <<<cdivll((long long)Etot * 4, 256), 256, 0, stream>>>(ei, E, Etot, mMax, eV, sSum);
    k_edge_alpha<<<cdivll((long long)Etot * 4, 256), 256, 0, stream>>>(ei, E, Etot, sSum, eV);
    k_edge_msg  <<<cdivll((long long)Etot * 256, 256), 256, 0, stream>>>(ei, E, (long long)Etot * 256, eV, hp, acc);
    k_finalize  <<<cdivll((long long)N * 64, 256), 256, 0, stream>>>(acc, bl[L], h, N);
  }

  k_fill<<<cdivll(64 * 64, 256), 256, 0, stream>>>(gs, 0.f, 64 * 64);
  k_fill<<<1, 64, 0, stream>>>(gc, 0.f, 64);
  k_pool<<<cdivll((long long)N * 64, 256), 256, 0, stream>>>(h, batch, gs, gc, N);
  k_pool_div<<<16, 256, 0, stream>>>(gs, gc);
  k_mlp<<<1, 1024, 0, stream>>>(gs, mW1, mb1, mW2, mb2, rW, rb, (float*)d_out);
}